// DistancePenaltyLoss_15333033246933
// MI455X (gfx1250) — compile-verified
//
#include <hip/hip_runtime.h>
#include <hip/hip_bf16.h>
#include <stdint.h>

#define TPB 256
#define C_NODES 22
#define A_AREAS 6
#define ROWS_PER_TILE 512
#define F4_PER_TILE (ROWS_PER_TILE * C_NODES / 4)   /* 2816 float4 per tile */
#define F4_PER_THREAD (F4_PER_TILE / TPB)           /* 11 float4 per thread */

#if __has_builtin(__builtin_amdgcn_s_wait_asynccnt)
#define WAIT_ASYNCCNT(n) __builtin_amdgcn_s_wait_asynccnt(n)
#else
#define WAIT_ASYNCCNT(n) asm volatile("s_wait_asynccnt %0" ::"i"(n) : "memory")
#endif

// ---------------------------------------------------------------------------
// Main kernel: double-buffered GLOBAL_LOAD_ASYNC_TO_LDS_B128 pipeline.
// Compute on buf[cb] overlaps the async fill of buf[cb^1] (ASYNCcnt-tracked).
// ---------------------------------------------------------------------------
__global__ __launch_bounds__(TPB) void dploss_main(
    const float* __restrict__ logits,
    const int*   __restrict__ targets,
    const float* __restrict__ dnode,
    const float* __restrict__ darea,
    const int*   __restrict__ n2a,
    float*       __restrict__ partials,
    int nrows, int ntiles)
{
    __shared__ float Msh[C_NODES * C_NODES];            // combined penalty matrix
    __shared__ float buf[2][ROWS_PER_TILE * C_NODES];   // 2 x 45056 B tiles
    __shared__ float red[TPB];

    const int tid = threadIdx.x;
    const int total_f4 = (nrows >> 1) * (C_NODES >> 1); // exact: B even, C=22
    const unsigned long long gbase = (unsigned long long)(uintptr_t)logits;

    // Issue exactly 11 async b128 copies per wave for tile `t` into buf[b].
    // Tail handling via index CLAMP (uniform instruction count -> ASYNCcnt
    // discipline holds; duplicated 16B reads on the tail are harmless).
    auto stage = [&](int t, int b) {
        const long long f4base = (long long)t * F4_PER_TILE;
#pragma unroll
        for (int i = 0; i < F4_PER_THREAD; ++i) {
            const int slot = i * TPB + tid;
            long long gi = f4base + slot;
            if (gi >= total_f4) gi = total_f4 - 1;
            const unsigned voff = (unsigned)(gi << 4);                      // byte offset
            const unsigned lds  = (unsigned)(uintptr_t)&buf[b][4 * slot];   // AS3 offset
            asm volatile("global_load_async_to_lds_b128 %0, %1, %2"
                         :: "v"(lds), "v"(voff), "s"(gbase) : "memory");
        }
    };

    // Prologue: start DMA for this block's first tile ASAP.
    int tile = blockIdx.x;
    int cb = 0;
    if (tile < ntiles) stage(tile, cb);

    // Build M[t][j] = D_node[t][j] + D_area[n2a[t]][n2a[j]] while DMA runs.
    for (int idx = tid; idx < C_NODES * C_NODES; idx += TPB) {
        const int t = idx / C_NODES;
        const int j = idx - t * C_NODES;
        Msh[idx] = dnode[idx] + darea[n2a[t] * A_AREAS + n2a[j]];
    }

    float acc = 0.0f;

    for (; tile < ntiles; tile += gridDim.x) {
        // Issue this iteration's target-index loads FIRST: their HBM latency
        // hides behind the async wait + barrier below (clamped, guarded later).
        const int rowbase = tile * ROWS_PER_TILE + 2 * tid;
        const int cr0 = (rowbase     < nrows) ? rowbase     : (nrows - 1);
        const int cr1 = (rowbase + 1 < nrows) ? rowbase + 1 : (nrows - 1);
        const int t0 = targets[cr0];
        const int t1 = targets[cr1];

        // Kick off the next tile into the other buffer (dummy clamped stage on
        // the last iteration keeps the per-wave ASYNCcnt bookkeeping uniform).
        const int nt = tile + (int)gridDim.x;
        const int nb = cb ^ 1;
        stage(nt < ntiles ? nt : tile, nb);

        // <=11 outstanding  ==>  the 11 copies filling buf[cb] are complete
        // for this wave (async loads complete in order) ...
        WAIT_ASYNCCNT(11);
        __syncthreads();   // ... and for every wave in the workgroup.

        // Each thread owns 2 consecutive rows of the tile.
#pragma unroll
        for (int r = 0; r < 2; ++r) {
            const int row = rowbase + r;
            if (row < nrows) {
                const int   t  = (r == 0) ? t0 : t1;
                const float* __restrict__ lr = &buf[cb][(2 * tid + r) * C_NODES];
                float v[C_NODES];
                float mx = -3.402823466e38f;
#pragma unroll
                for (int j = 0; j < C_NODES; ++j) {
                    v[j] = lr[j];
                    mx = fmaxf(mx, v[j]);
                }
                const float logit_t = lr[t];                 // dynamic idx via LDS
                const float* __restrict__ mr = Msh + t * C_NODES;
                float s = 0.0f, dot = 0.0f;
#pragma unroll
                for (int j = 0; j < C_NODES; ++j) {
                    const float e = __expf(v[j] - mx);       // v_exp_f32 fast path
                    s += e;
                    dot = __fmaf_rn(e, mr[j], dot);
                }
                const float logZ = mx + __logf(s);
                const float inv  = __builtin_amdgcn_rcpf(s); // v_rcp_f32 (1 ULP)
                acc += (logZ - logit_t) + dot * inv;         // CE + node + area pen
            }
        }
        __syncthreads();   // all reads of buf[cb] done before it is refilled
        cb = nb;
    }

    // Block tree reduction (wave32-safe plain LDS tree).
    red[tid] = acc;
    __syncthreads();
#pragma unroll
    for (int off = TPB / 2; off > 0; off >>= 1) {
        if (tid < off) red[tid] += red[tid + off];
        __syncthreads();
    }
    if (tid == 0) partials[blockIdx.x] = red[0];
}

// ---------------------------------------------------------------------------
// Deterministic final reduction (no float atomics -> bit-stable replays).
// ---------------------------------------------------------------------------
__global__ __launch_bounds__(TPB) void dploss_reduce(
    const float* __restrict__ partials, int n, float inv_b,
    float* __restrict__ out)
{
    __shared__ float red[TPB];
    float a = 0.0f;
    for (int i = threadIdx.x; i < n; i += TPB) a += partials[i];
    red[threadIdx.x] = a;
    __syncthreads();
#pragma unroll
    for (int off = TPB / 2; off > 0; off >>= 1) {
        if (threadIdx.x < off) red[threadIdx.x] += red[threadIdx.x + off];
        __syncthreads();
    }
    if (threadIdx.x == 0) out[0] = red[0] * inv_b;
}

extern "C" void kernel_launch(void* const* d_in, const int* in_sizes, int n_in,
                              void* d_out, int out_size, void* d_ws, size_t ws_size,
                              hipStream_t stream) {
    const float* logits  = (const float*)d_in[0];
    const int*   targets = (const int*)  d_in[1];
    const float* dnode   = (const float*)d_in[2];
    const float* darea   = (const float*)d_in[3];
    const int*   n2a     = (const int*)  d_in[4];
    float*       out     = (float*)d_out;

    const int nrows  = in_sizes[1];                         // B = 2,000,000
    const int ntiles = (nrows + ROWS_PER_TILE - 1) / ROWS_PER_TILE;
    int grid = 1024;
    if (grid > ntiles) grid = ntiles;

    float* partials = (float*)d_ws;                         // grid * 4 bytes

    dploss_main<<<grid, TPB, 0, stream>>>(logits, targets, dnode, darea, n2a,
                                          partials, nrows, ntiles);
    dploss_reduce<<<1, TPB, 0, stream>>>(partials, grid, 1.0f / (float)nrows, out);
}